// Seq2seq_12146167513166
// MI455X (gfx1250) — compile-verified
//
#include <hip/hip_runtime.h>
#include <math.h>

typedef float v2f __attribute__((ext_vector_type(2)));
typedef float v8f __attribute__((ext_vector_type(8)));

#define VOCAB 100000
#define EMB   100
#define HID   100
#define CTX   60
#define FACT  20
#define DESC  60
#define KDIM  100           // = HID = EMB
#define KC    25            // KDIM / 4
#define MT    4             // M tiles (60 rows padded to 64)
#define NTILES 6250         // VOCAB / 16
#define WAVES 4             // waves per block in GEMM kernel
#define TPW   10            // N-tiles per wave
#define GBLOCKS 157         // ceil(6250 / (WAVES*TPW))
#define NEG_BIG (-1.0e30f)  // finite "-inf" sentinel (avoids inf-inf NaNs)

// workspace layout (float offsets)
#define OFF_H0   0                          // 100
#define OFF_GI   100                        // 60*300 = 18000
#define OFF_H    (100 + DESC*3*HID)         // 18100 : padded H [64][100]
#define OFF_PMAX (OFF_H + 64*KDIM)          // 24500 : [GBLOCKS][60]
#define OFF_PSUM (OFF_PMAX + GBLOCKS*DESC)  // 33920 : [GBLOCKS][60]

// ---------------------------------------------------------------------------
// 1) facts (positional fact encoder) + h0 = facts_flat @ w1.T + b1
// ---------------------------------------------------------------------------
__global__ void k_facts_h0(const int* __restrict__ ctx,
                           const float* __restrict__ emb_ctx,
                           const float* __restrict__ w1,
                           const float* __restrict__ b1,
                           float* __restrict__ ws) {
  __shared__ float sF[CTX * EMB];
  const int tid = threadIdx.x;
  for (int idx = tid; idx < CTX * EMB; idx += blockDim.x) {
    const int c = idx / EMB, e = idx % EMB;
    const float ep = (float)e / (float)(EMB - 1);
    float acc = 0.f;
#pragma unroll
    for (int s = 0; s < FACT; ++s) {
      const float sp = (float)s / (float)(FACT - 1);
      const float l = 1.f - sp - ep * (1.f - 2.f * sp);
      const int tok = ctx[c * FACT + s];
      acc += emb_ctx[(size_t)tok * EMB + e] * l;
    }
    sF[idx] = acc;
  }
  __syncthreads();
  if (tid < HID) {
    const float* wr = w1 + (size_t)tid * (CTX * EMB);
    float h = b1[tid];
    for (int i = 0; i < CTX * EMB; ++i) h += sF[i] * wr[i];
    ws[OFF_H0 + tid] = h;
  }
}

// ---------------------------------------------------------------------------
// 2) gi[t] = emb_dec[x_t] @ w_ih.T + b_ih  (teacher forcing -> parallel in t)
// ---------------------------------------------------------------------------
__global__ void k_gi(const int* __restrict__ desc,
                     const float* __restrict__ emb_dec,
                     const float* __restrict__ w_ih,
                     const float* __restrict__ b_ih,
                     float* __restrict__ ws) {
  __shared__ float sx[EMB];
  const int t = blockIdx.x;
  const int tok = (t == 0) ? 1 : desc[t - 1];   // SOS = token 1
  for (int i = threadIdx.x; i < EMB; i += blockDim.x)
    sx[i] = emb_dec[(size_t)tok * EMB + i];
  __syncthreads();
  const int j = threadIdx.x;
  if (j < 3 * HID) {
    const float* wr = w_ih + (size_t)j * EMB;
    float g = b_ih[j];
    for (int k = 0; k < EMB; ++k) g += sx[k] * wr[k];
    ws[OFF_GI + t * (3 * HID) + j] = g;
  }
}

// ---------------------------------------------------------------------------
// 3) sequential GRU recurrence; writes zero-padded H[64][100]
// ---------------------------------------------------------------------------
__global__ void k_gru(const float* __restrict__ w_hh,
                      const float* __restrict__ b_hh,
                      float* __restrict__ ws) {
  __shared__ float h[HID];
  __shared__ float gh[3 * HID];
  const int tid = threadIdx.x;
  if (tid < HID) h[tid] = ws[OFF_H0 + tid];
  for (int i = tid; i < 4 * KDIM; i += blockDim.x)   // zero pad rows 60..63
    ws[OFF_H + 60 * KDIM + i] = 0.f;
  __syncthreads();
  for (int t = 0; t < DESC; ++t) {
    if (tid < 3 * HID) {
      const float* wr = w_hh + (size_t)tid * HID;
      float g = b_hh[tid];
      for (int k = 0; k < HID; ++k) g += h[k] * wr[k];
      gh[tid] = g;
    }
    __syncthreads();
    if (tid < HID) {
      const float* gi = ws + OFF_GI + t * (3 * HID);
      const float r = 1.f / (1.f + __expf(-(gi[tid]           + gh[tid])));
      const float z = 1.f / (1.f + __expf(-(gi[HID + tid]     + gh[HID + tid])));
      const float n = tanhf(gi[2 * HID + tid] + r * gh[2 * HID + tid]);
      const float hn = (1.f - z) * n + z * h[tid];
      h[tid] = hn;
      ws[OFF_H + t * KDIM + tid] = hn;
    }
    __syncthreads();
  }
}

// ---------------------------------------------------------------------------
// 4) WMMA GEMM  [64,100] x [100,100000]  fused with online softmax partials
//    (per-lane register-resident online logsumexp; one cross-lane merge/kernel;
//     target-token logits are computed separately in k_loss -> hot loop stays
//     branchless with full EXEC for WMMA)
// ---------------------------------------------------------------------------
__global__ void __launch_bounds__(128)
k_logits(const float* __restrict__ w_out,
         const float* __restrict__ b_out,
         float* __restrict__ ws) {
  __shared__ __align__(16) float sH[64][104];            // padded H, stride 104
  __shared__ __align__(16) float sW[WAVES][16][104];     // per-wave w_out tile
  __shared__ float sMax[WAVES][DESC];
  __shared__ float sSum[WAVES][DESC];

  const int tid  = threadIdx.x;
  const int wave = tid >> 5;
  const int lane = tid & 31;
  const int half = lane >> 4;   // 0: lanes 0-15, 1: lanes 16-31
  const int l16  = lane & 15;

  // stage padded H into LDS (shared by all 4 waves)
  for (int i = tid; i < 64 * 104; i += 128) {
    const int r = i / 104, c = i % 104;
    sH[r][c] = (c < KDIM) ? ws[OFF_H + r * KDIM + c] : 0.f;
  }
  __syncthreads();

  // register-resident per-lane online (M,S) state
  float rM[MT][8], rS[MT][8];
#pragma unroll
  for (int m = 0; m < MT; ++m)
#pragma unroll
    for (int v = 0; v < 8; ++v) { rM[m][v] = NEG_BIG; rS[m][v] = 0.f; }

  const int tile0 = blockIdx.x * (WAVES * TPW) + wave * TPW;
  for (int ti = 0; ti < TPW; ++ti) {
    const int tile = tile0 + ti;
    if (tile >= NTILES) break;               // wave-uniform
    const int n0 = tile * 16;

    if (tile + 1 < NTILES)                   // global_prefetch next w_out tile
      __builtin_prefetch(w_out + (size_t)(n0 + 16) * KDIM, 0, 1);

    // per-wave cooperative load: 16 rows x 100 floats of w_out (float4)
    for (int u = lane; u < 16 * KC; u += 32) {
      const int r = u / KC, q = u % KC;
      const float4 v = ((const float4*)(w_out + (size_t)(n0 + r) * KDIM))[q];
      *(float4*)&sW[wave][r][q * 4] = v;
    }
    asm volatile("s_wait_dscnt 0" ::: "memory");  // cross-lane LDS handoff (wave-local)

    v8f acc[MT];
    const v8f vzero = {0.f, 0.f, 0.f, 0.f, 0.f, 0.f, 0.f, 0.f};
#pragma unroll
    for (int m = 0; m < MT; ++m) acc[m] = vzero;

#pragma unroll 5
    for (int kc = 0; kc < KC; ++kc) {
      const int kb = kc * 4 + half * 2;
      // B frag (4x16): VGPR0 = K{0,2}, VGPR1 = K{1,3}; N = lane&15
      const v2f b = *(const v2f*)&sW[wave][l16][kb];
#pragma unroll
      for (int m = 0; m < MT; ++m) {
        // A frag (16x4): lanes 0-15 -> K{0,1}, lanes 16-31 -> K{2,3}; M = lane&15
        const v2f a = *(const v2f*)&sH[m * 16 + l16][kb];
        acc[m] = __builtin_amdgcn_wmma_f32_16x16x4_f32(
            false, a, false, b, (short)0, acc[m], false, false);
      }
    }

    // epilogue: branchless per-lane online logsumexp update
    const float bo = b_out[n0 + l16];
#pragma unroll
    for (int m = 0; m < MT; ++m) {
#pragma unroll
      for (int v = 0; v < 8; ++v) {
        const float x  = acc[m][v] + bo;
        const float Mn = fmaxf(rM[m][v], x);
        rS[m][v] = rS[m][v] * __expf(rM[m][v] - Mn) + __expf(x - Mn);
        rM[m][v] = Mn;
      }
    }
  }

  // one cross-lane (16-lane) logsumexp merge per value, then publish per wave
#pragma unroll
  for (int m = 0; m < MT; ++m) {
#pragma unroll
    for (int v = 0; v < 8; ++v) {
      float M = rM[m][v], S = rS[m][v];
#pragma unroll
      for (int off = 8; off >= 1; off >>= 1) {
        const float M2 = __shfl_xor(M, off, 16);
        const float S2 = __shfl_xor(S, off, 16);
        const float Mn = fmaxf(M, M2);
        S = S * __expf(M - Mn) + S2 * __expf(M2 - Mn);
        M = Mn;
      }
      const int row = m * 16 + v + half * 8;
      if (l16 == 0 && row < DESC) {   // lanes 0 & 16 write distinct rows
        sMax[wave][row] = M;
        sSum[wave][row] = S;
      }
    }
  }

  __syncthreads();
  if (tid < DESC) {   // merge 4 waves -> per-block partial (max, sumexp)
    float M = sMax[0][tid], S = sSum[0][tid];
#pragma unroll
    for (int w = 1; w < WAVES; ++w) {
      const float m2 = sMax[w][tid], s2 = sSum[w][tid];
      const float Mn = fmaxf(M, m2);
      S = S * __expf(M - Mn) + s2 * __expf(m2 - Mn);
      M = Mn;
    }
    ws[OFF_PMAX + blockIdx.x * DESC + tid] = M;
    ws[OFF_PSUM + blockIdx.x * DESC + tid] = S;
  }
}

// ---------------------------------------------------------------------------
// 5) loss: per-row target logit (direct 100-length dot) + logsumexp reduction
// ---------------------------------------------------------------------------
__global__ void k_loss(const int* __restrict__ desc,
                       const float* __restrict__ w_out,
                       const float* __restrict__ b_out,
                       const float* __restrict__ ws,
                       float* __restrict__ out) {
  __shared__ float sL[64];
  const int t = threadIdx.x;
  float loss = 0.f;
  if (t < DESC) {
    // logit of the target token: h_t . w_out[desc[t]] + b_out[desc[t]]
    const int tok = desc[t];
    const float* hr = ws + OFF_H + t * KDIM;
    const float* wr = w_out + (size_t)tok * KDIM;
    float xtok = b_out[tok];
    for (int k = 0; k < KDIM; ++k) xtok += hr[k] * wr[k];

    // merge chunk partials into a global logsumexp
    float M = NEG_BIG, S = 0.f;
    for (int c = 0; c < GBLOCKS; ++c) {
      const float m2 = ws[OFF_PMAX + c * DESC + t];
      const float s2 = ws[OFF_PSUM + c * DESC + t];
      const float Mn = fmaxf(M, m2);
      S = S * expf(M - Mn) + s2 * expf(m2 - Mn);
      M = Mn;
    }
    loss = (logf(S) + M) - xtok;
  }
  sL[t] = loss;
  __syncthreads();
  for (int s = 32; s > 0; s >>= 1) {
    if (t < s) sL[t] += sL[t + s];
    __syncthreads();
  }
  if (t == 0) out[0] = sL[0];
}

// ---------------------------------------------------------------------------
extern "C" void kernel_launch(void* const* d_in, const int* in_sizes, int n_in,
                              void* d_out, int out_size, void* d_ws, size_t ws_size,
                              hipStream_t stream) {
  const int*   ctx     = (const int*)d_in[0];
  // d_in[1] fact_lengths: unused by the reference forward
  const int*   desc    = (const int*)d_in[2];
  const float* emb_ctx = (const float*)d_in[3];
  const float* emb_dec = (const float*)d_in[4];
  const float* w1      = (const float*)d_in[5];
  const float* b1      = (const float*)d_in[6];
  const float* w_ih    = (const float*)d_in[7];
  const float* w_hh    = (const float*)d_in[8];
  const float* b_ih    = (const float*)d_in[9];
  const float* b_hh    = (const float*)d_in[10];
  const float* w_out   = (const float*)d_in[11];
  const float* b_out   = (const float*)d_in[12];
  float* ws  = (float*)d_ws;
  float* out = (float*)d_out;

  k_facts_h0<<<1, 128, 0, stream>>>(ctx, emb_ctx, w1, b1, ws);
  k_gi<<<DESC, 320, 0, stream>>>(desc, emb_dec, w_ih, b_ih, ws);
  k_gru<<<1, 320, 0, stream>>>(w_hh, b_hh, ws);
  k_logits<<<GBLOCKS, 128, 0, stream>>>(w_out, b_out, ws);
  k_loss<<<1, 64, 0, stream>>>(desc, w_out, b_out, ws, out);
}